// MaskDE_5428838662291
// MI455X (gfx1250) — compile-verified
//
#include <hip/hip_runtime.h>
#include <stdint.h>

typedef float        v4f __attribute__((ext_vector_type(4)));
typedef unsigned int v4u __attribute__((ext_vector_type(4)));

#define AS1 __attribute__((address_space(1)))
#define AS3 __attribute__((address_space(3)))

#if __has_builtin(__builtin_amdgcn_global_load_async_to_lds_b32) && \
    __has_builtin(__builtin_amdgcn_s_wait_asynccnt)
#define USE_ASYNC_LDS 1
#else
#define USE_ASYNC_LDS 0
#endif

// ---------------------------------------------------------------------------
// Kernel 1: build idx[] = positions where mask is true.
// Robust to the mask being stored as 1-byte bools or 4-byte ints: we know the
// expected count m (derived on host from out_size), so pick the interpretation
// whose popcount matches.
// ---------------------------------------------------------------------------
__global__ void MaskDE_build_idx(const void* __restrict__ maskp, int D, int m,
                                 int* __restrict__ idx) {
    if (threadIdx.x != 0 || blockIdx.x != 0) return;
    const unsigned char* mb = (const unsigned char*)maskp;
    int cb = 0;
    for (int d = 0; d < D; ++d) cb += (mb[d] != 0);
    if (cb == m) {
        int c = 0;
        for (int d = 0; d < D; ++d) if (mb[d]) idx[c++] = d;
    } else {
        const int* mi = (const int*)maskp;
        int c = 0;
        for (int d = 0; d < D; ++d) if (mi[d]) idx[c++] = d;
    }
}

// ---------------------------------------------------------------------------
// Kernel 2: build packed combination table (one u32 per output feature).
// code = a | b<<8 | c<<16 with local indices into xm; unused slots = m
// (sentinel slot holds 1.0f). Layout matches the reference concatenation:
// [order-1 (m)], [order-2 (m(m+1)/2)], [order-3 (m(m+1)(m+2)/6)], each in
// itertools.combinations_with_replacement lexicographic order.
// One block per first-index i; closed-form tetrahedral offsets.
// ---------------------------------------------------------------------------
__global__ void MaskDE_build_table(uint32_t* __restrict__ table, int m) {
    const int i   = blockIdx.x;     // 0..m-1
    const int tid = threadIdx.x;
    const uint32_t mm = (uint32_t)m;

    // order-1
    if (tid == 0) table[i] = (uint32_t)i | (mm << 8) | (mm << 16);

    // order-2: pairs (i,j), j >= i
    const int off2 = m;
    const int r2   = i * m - (i * (i - 1)) / 2;   // pairs with first index < i
    for (int j = i + tid; j < m; j += blockDim.x)
        table[off2 + r2 + (j - i)] =
            (uint32_t)i | ((uint32_t)j << 8) | (mm << 16);

    // order-3: triples (i,j,k), i<=j<=k
    const long T3m  = (long)m * (m + 1) * (m + 2) / 6;
    const int  off3 = m + m * (m + 1) / 2;
    const int  n    = m - i;                           // remaining elements
    const long t3i  = T3m - (long)n * (n + 1) * (n + 2) / 6; // triples, first<i
    for (int j = i + tid; j < m; j += blockDim.x) {
        const int  jp   = j - i;
        const long base = off3 + t3i + (long)jp * n - (long)jp * (jp - 1) / 2;
        const uint32_t code = (uint32_t)i | ((uint32_t)j << 8);
        for (int k = j; k < m; ++k)
            table[base + (k - j)] = code | ((uint32_t)k << 16);
    }
}

// ---------------------------------------------------------------------------
// Kernel 3: main expansion. 8 rows per block; xm staged in LDS via CDNA5
// async global->LDS loads; table read once per 8 rows as uint4; outputs
// written as non-temporal float4 streams (store-bandwidth bound).
// ---------------------------------------------------------------------------
__device__ __forceinline__ float MaskDE_prod3(const float* sx, unsigned c) {
    return sx[c & 255u] * sx[(c >> 8) & 255u] * sx[(c >> 16) & 255u];
}

__global__ __launch_bounds__(256)
void MaskDE_expand(const float* __restrict__ x, const int* __restrict__ idx,
                   const uint32_t* __restrict__ table, float* __restrict__ out,
                   int m, int F, int B, int D) {
    constexpr int RPB = 8;
    __shared__ float s_xm[RPB][88];   // m <= 80; slot [m] = 1.0f sentinel

    const int tid   = threadIdx.x;
    const int row0  = blockIdx.x * RPB;
    const int nrows = (B - row0 < RPB) ? (B - row0) : RPB;

    // Gather masked columns of each row into LDS.
    if (tid < m) {
        const int col = idx[tid];
        for (int r = 0; r < nrows; ++r) {
            const float* src = x + (size_t)(row0 + r) * (size_t)D + col;
#if USE_ASYNC_LDS
            AS1 int* gsrc = (AS1 int*)(uintptr_t)src;
            AS3 int* ldst = (AS3 int*)(uint32_t)(uintptr_t)&s_xm[r][tid];
            __builtin_amdgcn_global_load_async_to_lds_b32(gsrc, ldst, 0, 0);
#else
            s_xm[r][tid] = *src;
#endif
        }
    }
    if (tid < RPB) s_xm[tid][m] = 1.0f;
#if USE_ASYNC_LDS
    __builtin_amdgcn_s_wait_asynccnt(0);
#endif
    __syncthreads();

    const size_t ob = (size_t)row0 * (size_t)F;

    if ((F & 3) == 0 && nrows == RPB) {
        // Fast path: uint4 table reads, float4 non-temporal stores, 8 rows.
        const v4u* __restrict__ t4 = (const v4u*)table;
        const int F4 = F >> 2;
        for (int g = tid; g < F4; g += 256) {
            __builtin_prefetch(t4 + g + 256, 0, 0);   // global_prefetch_b8
            const v4u c = t4[g];
#pragma unroll
            for (int r = 0; r < RPB; ++r) {
                const float* sx = s_xm[r];
                v4f v;
                v.x = MaskDE_prod3(sx, c.x);
                v.y = MaskDE_prod3(sx, c.y);
                v.z = MaskDE_prod3(sx, c.z);
                v.w = MaskDE_prod3(sx, c.w);
                __builtin_nontemporal_store(
                    v, (v4f*)(out + ob + (size_t)r * F + ((size_t)g << 2)));
            }
        }
    } else {
        // General path: scalar, still lane-coalesced.
        for (int p = tid; p < F; p += 256) {
            const unsigned c = table[p];
            for (int r = 0; r < nrows; ++r)
                __builtin_nontemporal_store(
                    MaskDE_prod3(s_xm[r], c), out + ob + (size_t)r * F + p);
        }
    }
}

// ---------------------------------------------------------------------------
extern "C" void kernel_launch(void* const* d_in, const int* in_sizes, int n_in,
                              void* d_out, int out_size, void* d_ws, size_t ws_size,
                              hipStream_t stream) {
    const float* x    = (const float*)d_in[0];
    const void*  mask = (const void*)d_in[1];
    const int D = in_sizes[1];                 // 80
    const int B = in_sizes[0] / D;             // 8192
    const long F = (B > 0) ? (long)out_size / B : 0;

    // Invert F(m) = m + m(m+1)/2 + m(m+1)(m+2)/6  (strictly increasing in m).
    int m = -1;
    for (int t = 0; t <= D; ++t) {
        const long f = (long)t + (long)t * (t + 1) / 2
                     + (long)t * (t + 1) * (t + 2) / 6;
        if (f == F) { m = t; break; }
    }
    if (m < 0) m = 0;   // should not happen; avoid UB

    int*      idx   = (int*)d_ws;
    uint32_t* table = (uint32_t*)((char*)d_ws + 512);   // 16B-aligned

    MaskDE_build_idx<<<1, 1, 0, stream>>>(mask, D, m, idx);
    if (m > 0 && B > 0) {
        MaskDE_build_table<<<m, 128, 0, stream>>>(table, m);
        const int grid = (B + 7) / 8;
        MaskDE_expand<<<grid, 256, 0, stream>>>(x, idx, table, (float*)d_out,
                                                m, (int)F, B, D);
    }
}